// GCNConv_69080253988962
// MI455X (gfx1250) — compile-verified
//
#include <hip/hip_runtime.h>

typedef __attribute__((ext_vector_type(2))) float v2f;
typedef __attribute__((ext_vector_type(8))) float v8f;

#define N_NODES 50000
#define N_EDGES 1250000
#define D_FEAT  64

// Fused GCN conv: segment-mean (sorted ids, binary-searched ranges, no atomics)
// + residual + dense(64x64) via V_WMMA_F32_16X16X4_F32.
// One workgroup (4 wave32) per 16-node row tile; 50000 = 3125 * 16 exactly.
//
// Phase 2 is the bandwidth phase (320 MB dst stream): b128 loads, 2 edge rows
// per wave instruction (512 B contiguous), unrolled x4 (8 edges / iter) so each
// wave keeps 4 independent b128 loads (2 KB) in flight -> saturates HBM BDP.
__global__ __launch_bounds__(128) void gcn_fused_kernel(
    const float* __restrict__ src,       // [N_NODES, 64]
    const float* __restrict__ dst,       // [N_EDGES, 64]
    const int*   __restrict__ seg,       // [N_EDGES] sorted
    const float* __restrict__ W,         // [64, 64]
    const float* __restrict__ bias,      // [64]
    float*       __restrict__ out)       // [N_NODES, 64]
{
    __shared__ float xs[16][D_FEAT];     // x = mean + src tile (4 KB)
    __shared__ int   ls[17];             // edge-range boundaries for the 16 nodes

    const int tid  = threadIdx.x;
    const int wave = tid >> 5;
    const int lane = tid & 31;
    const int half = lane >> 4;          // 0: lanes 0-15, 1: lanes 16-31
    const int l15  = lane & 15;
    const int base = blockIdx.x << 4;    // first node of this tile

    // ---- Phase 1: lower_bound(seg, base + t) for t = 0..16 ----
    if (tid < 17) {
        const int v = base + tid;
        int lo = 0, hi = N_EDGES;
        while (lo < hi) {
            const int mid = (lo + hi) >> 1;
            if (seg[mid] < v) lo = mid + 1; else hi = mid;
        }
        ls[tid] = lo;
    }
    __syncthreads();

    // ---- Phase 2: segment mean + residual -> LDS ----
    // Lane-half h owns edges of parity h; lane covers features l15*4..l15*4+3.
    // One b128 load per lane => wave covers edges e and e+1 (512 B contiguous).
    const float4* __restrict__ dst4 = (const float4*)dst;
    const float4* __restrict__ src4 = (const float4*)src;
    for (int j = 0; j < 4; ++j) {
        const int n  = wave * 4 + j;         // node within tile
        const int s0 = ls[n];
        const int s1 = ls[n + 1];
        const int cnt = s1 - s0;

        float4 a0 = make_float4(0.f, 0.f, 0.f, 0.f);
        float4 a1 = a0, a2 = a0, a3 = a0;

        int e = s0;
        // 8 edges per iteration: 4 independent b128 loads in flight per wave
        for (; e + 8 <= s1; e += 8) {
            const float4 d0 = dst4[(size_t)(e + 0 + half) * 16 + l15];
            const float4 d1 = dst4[(size_t)(e + 2 + half) * 16 + l15];
            const float4 d2 = dst4[(size_t)(e + 4 + half) * 16 + l15];
            const float4 d3 = dst4[(size_t)(e + 6 + half) * 16 + l15];
            a0.x += d0.x; a0.y += d0.y; a0.z += d0.z; a0.w += d0.w;
            a1.x += d1.x; a1.y += d1.y; a1.z += d1.z; a1.w += d1.w;
            a2.x += d2.x; a2.y += d2.y; a2.z += d2.z; a2.w += d2.w;
            a3.x += d3.x; a3.y += d3.y; a3.z += d3.z; a3.w += d3.w;
        }
        // pair tail
        for (; e + 2 <= s1; e += 2) {
            const float4 d0 = dst4[(size_t)(e + half) * 16 + l15];
            a0.x += d0.x; a0.y += d0.y; a0.z += d0.z; a0.w += d0.w;
        }
        // odd single-edge tail: only lane-half 0 contributes
        if (e < s1 && half == 0) {
            const float4 d0 = dst4[(size_t)e * 16 + l15];
            a0.x += d0.x; a0.y += d0.y; a0.z += d0.z; a0.w += d0.w;
        }

        // merge the 4 accumulators, then merge even/odd halves across lanes
        float4 t;
        t.x = (a0.x + a1.x) + (a2.x + a3.x);
        t.y = (a0.y + a1.y) + (a2.y + a3.y);
        t.z = (a0.z + a1.z) + (a2.z + a3.z);
        t.w = (a0.w + a1.w) + (a2.w + a3.w);
        t.x += __shfl_xor(t.x, 16, 32);
        t.y += __shfl_xor(t.y, 16, 32);
        t.z += __shfl_xor(t.z, 16, 32);
        t.w += __shfl_xor(t.w, 16, 32);

        if (lane < 16) {
            const float  inv = 1.0f / (float)(cnt > 0 ? cnt : 1);
            const float4 s   = src4[(size_t)(base + n) * 16 + l15];
            float4 x;
            x.x = t.x * inv + s.x;
            x.y = t.y * inv + s.y;
            x.z = t.z * inv + s.z;
            x.w = t.w * inv + s.w;
            *(float4*)&xs[n][4 * l15] = x;   // 16B-aligned LDS store
        }
    }
    __syncthreads();

    // ---- Phase 3: D[16x16] = X[16x64] @ W[64, 16*wave..] via f32 WMMA ----
    // A 16x4 f32 layout: lanes 0-15 -> K=0,1 ; lanes 16-31 -> K=2,3 (M = lane&15)
    // B 4x16 f32 layout: VGPR0/1 -> rows K(+0,+1); lane half selects K(+2,+3)
    const int colBase = wave << 4;       // 16 output cols per wave

    v8f c = {0.f, 0.f, 0.f, 0.f, 0.f, 0.f, 0.f, 0.f};
#pragma unroll
    for (int k0 = 0; k0 < 16; ++k0) {
        const int kb = k0 * 4 + half * 2;
        v2f a, b;
        a.x = xs[l15][kb];
        a.y = xs[l15][kb + 1];
        b.x = W[(size_t)kb       * D_FEAT + colBase + l15];
        b.y = W[(size_t)(kb + 1) * D_FEAT + colBase + l15];
        // (neg_a, A, neg_b, B, c_mod, C, reuse_a, reuse_b)
        c = __builtin_amdgcn_wmma_f32_16x16x4_f32(
                false, a, false, b, (short)0, c, false, false);
    }

    // C/D layout: VGPR v -> row v (lanes 0-15) or v+8 (lanes 16-31), col = l15
    const float bv = bias[colBase + l15];
#pragma unroll
    for (int v = 0; v < 8; ++v) {
        const int row = v + half * 8;
        out[(size_t)(base + row) * D_FEAT + colBase + l15] = c[v] + bv;
    }
}

extern "C" void kernel_launch(void* const* d_in, const int* in_sizes, int n_in,
                              void* d_out, int out_size, void* d_ws, size_t ws_size,
                              hipStream_t stream) {
    // setup_inputs order: src, edge (unused), dst, segment_ids, W, b
    const float* src  = (const float*)d_in[0];
    const float* dst  = (const float*)d_in[2];
    const int*   seg  = (const int*)  d_in[3];
    const float* W    = (const float*)d_in[4];
    const float* bias = (const float*)d_in[5];
    float*       out  = (float*)d_out;

    (void)in_sizes; (void)n_in; (void)out_size; (void)d_ws; (void)ws_size;

    const int tiles = N_NODES / 16;      // 3125, exact
    gcn_fused_kernel<<<tiles, 128, 0, stream>>>(src, dst, seg, W, bias, out);
}